// OneLayerRNN_61933428408831
// MI455X (gfx1250) — compile-verified
//
#include <hip/hip_runtime.h>
#include <math.h>

// ---------------------------------------------------------------------------
// One-layer Elman RNN, B=8 P=32 S=2048 I=32 H=64 O=32  (all f32)
// d_out layout: obs[256*2048*32] | hidden[256*2048*64] | h_last[256*64]
// pre is computed in-place into the hidden region (same [bp][S][H] shape),
// then overwritten step-by-step by the scan kernel -> no workspace needed.
//
// Cache policy: x (read-once) and obs (write-once, never re-read) use
// non-temporal hints; pre/hidden (producer->consumer across kernels,
// 134 MiB < 192 MB L2) use default RT so they stay L2-resident.
// ---------------------------------------------------------------------------

typedef __attribute__((ext_vector_type(2))) float v2f;
typedef __attribute__((ext_vector_type(8))) float v8f;

__device__ __forceinline__ v8f wmma_f32_16x16x4(v2f a, v2f b, v8f c) {
  // D = A(16x4,f32) * B(4x16,f32) + C(16x16,f32)
  return __builtin_amdgcn_wmma_f32_16x16x4_f32(
      /*neg_a=*/false, a, /*neg_b=*/false, b,
      /*c_mod=*/(short)0, c, /*reuse_a=*/false, /*reuse_b=*/false);
}

// Hardware V_TANH_F32 (TRANS op) if the builtin is available; this sits on the
// 2048-step serial dependency chain, so latency here matters most.
__device__ __forceinline__ float fast_tanh(float v) {
#if __has_builtin(__builtin_amdgcn_tanhf)
  return __builtin_amdgcn_tanhf(v);
#elif __has_builtin(__builtin_amdgcn_tanh_f32)
  return __builtin_amdgcn_tanh_f32(v);
#else
  return tanhf(v);
#endif
}

// ---------------------------------------------------------------------------
// Kernel 1: pre[r][h] = x[r][:] . W_ih[h][:] + b_ih[h] + b_hh[h]
//   r = bp*S + s  (524288 rows), K=32, N=64. 4 waves/WG, 16 rows/wave.
// ---------------------------------------------------------------------------
__global__ __launch_bounds__(128) void pre_kernel(
    const float* __restrict__ x,     // [BP*S, 32]
    const float* __restrict__ W_ih,  // [64, 32]
    const float* __restrict__ b_ih,  // [64]
    const float* __restrict__ b_hh,  // [64]
    float* __restrict__ pre)         // [BP*S, 64]  (== hidden region of d_out)
{
  const int lane = threadIdx.x & 31;
  const int wave = threadIdx.x >> 5;
  const int m    = lane & 15;     // A row / B,C column-lane
  const int kh   = lane >> 4;     // K-half select (0/1)
  const long rowBase = (long)blockIdx.x * 64 + (long)wave * 16;

  // B = W_ih^T in WMMA B layout: 4 N-tiles x 8 K-chunks x 2 VGPRs
  v2f bmat[4][8];
#pragma unroll
  for (int t = 0; t < 4; ++t) {
    const int n = t * 16 + m;
#pragma unroll
    for (int kc = 0; kc < 8; ++kc) {
      const int k = kc * 4 + 2 * kh;
      bmat[t][kc].x = W_ih[n * 32 + k];
      bmat[t][kc].y = W_ih[n * 32 + k + 1];
    }
  }

  // A: 16 rows x K=32 of x (read-once stream -> non-temporal)
  v2f amat[8];
#pragma unroll
  for (int kc = 0; kc < 8; ++kc) {
    const int k = kc * 4 + 2 * kh;
    const float* xr = x + (rowBase + m) * 32 + k;
    amat[kc].x = __builtin_nontemporal_load(xr);
    amat[kc].y = __builtin_nontemporal_load(xr + 1);
  }

#pragma unroll
  for (int t = 0; t < 4; ++t) {
    const int n = t * 16 + m;
    const float bias = b_ih[n] + b_hh[n];
    v8f c;
#pragma unroll
    for (int j = 0; j < 8; ++j) c[j] = bias;   // bias folded into accumulator
#pragma unroll
    for (int kc = 0; kc < 8; ++kc)
      c = wmma_f32_16x16x4(amat[kc], bmat[t][kc], c);
#pragma unroll
    for (int j = 0; j < 8; ++j) {
      const long r = rowBase + j + 8 * kh;     // C/D: row = vgpr + 8*(lane>=16)
      pre[r * 64 + n] = c[j];                  // RT: re-read by scan kernel
    }
  }
}

// ---------------------------------------------------------------------------
// Kernel 2: sequential scan  h_t = tanh(pre_t + h_{t-1} @ W_hh^T)
//   16 WGs x 128 thr; WG g owns bp rows [16g,16g+16); wave w owns N-tile w.
//   h lives in LDS; pre is read from `hid` and overwritten with tanh result.
// ---------------------------------------------------------------------------
__global__ __launch_bounds__(128) void rnn_scan_kernel(
    const float* __restrict__ W_hh,  // [64, 64]
    float* __restrict__ hid,         // [256, 2048, 64]  in: pre, out: hidden
    float* __restrict__ h_last)      // [256, 64]
{
  __shared__ float hbuf[16 * 64];
  const int lane = threadIdx.x & 31;
  const int wave = threadIdx.x >> 5;   // N-tile (0..3)
  const int m    = lane & 15;
  const int kh   = lane >> 4;
  const int n    = wave * 16 + m;      // output column of this lane
  const int g    = blockIdx.x;         // row-group (0..15)
  const int S = 2048, H = 64;

  for (int i = threadIdx.x; i < 16 * 64; i += blockDim.x) hbuf[i] = 0.f;

  // B = W_hh^T, this wave's N-tile: 16 K-chunks x 2 VGPRs (stays in regs)
  v2f bmat[16];
#pragma unroll
  for (int kc = 0; kc < 16; ++kc) {
    const int k = kc * 4 + 2 * kh;
    bmat[kc].x = W_hh[n * H + k];
    bmat[kc].y = W_hh[n * H + k + 1];
  }
  __syncthreads();

  for (int s = 0; s < S; ++s) {
    // A = h_{s-1} rows from LDS
    v2f amat[16];
#pragma unroll
    for (int kc = 0; kc < 16; ++kc) {
      const int k = kc * 4 + 2 * kh;
      amat[kc].x = hbuf[m * H + k];
      amat[kc].y = hbuf[m * H + k + 1];
    }
    // C init = pre_t (resident in hid), coalesced 64B per row-half
    v8f c;
#pragma unroll
    for (int j = 0; j < 8; ++j) {
      const long r = (long)g * 16 + j + 8 * kh;
      c[j] = hid[(r * S + s) * H + n];
    }
    __syncthreads();  // all waves done reading h before it is overwritten

#pragma unroll
    for (int kc = 0; kc < 16; ++kc)
      c = wmma_f32_16x16x4(amat[kc], bmat[kc], c);
#pragma unroll
    for (int j = 0; j < 8; ++j) c[j] = fast_tanh(c[j]);

#pragma unroll
    for (int j = 0; j < 8; ++j) {
      const int  row = j + 8 * kh;
      const long r   = (long)g * 16 + row;
      hid[(r * S + s) * H + n] = c[j];   // hidden output (overwrites pre_t)
      hbuf[row * H + n]        = c[j];   // state for next step
    }
    __syncthreads();
  }

  // h_last = final h (hbuf is fully written; last barrier above ordered it)
#pragma unroll
  for (int j = 0; j < 8; ++j) {
    const int row = j + 8 * kh;
    h_last[((long)g * 16 + row) * H + n] = hbuf[row * H + n];
  }
}

// ---------------------------------------------------------------------------
// Kernel 3: obs[r][o] = hidden[r][:] . W_fc[o][:] + b_fc[o]   K=64, N=32
// ---------------------------------------------------------------------------
__global__ __launch_bounds__(128) void obs_kernel(
    const float* __restrict__ hid,   // [BP*S, 64]
    const float* __restrict__ W_fc,  // [32, 64]
    const float* __restrict__ b_fc,  // [32]
    float* __restrict__ obs)         // [BP*S, 32]
{
  const int lane = threadIdx.x & 31;
  const int wave = threadIdx.x >> 5;
  const int m    = lane & 15;
  const int kh   = lane >> 4;
  const long rowBase = (long)blockIdx.x * 64 + (long)wave * 16;
  const int H = 64;

  v2f bmat[2][16];
#pragma unroll
  for (int t = 0; t < 2; ++t) {
    const int n = t * 16 + m;
#pragma unroll
    for (int kc = 0; kc < 16; ++kc) {
      const int k = kc * 4 + 2 * kh;
      bmat[t][kc].x = W_fc[n * H + k];
      bmat[t][kc].y = W_fc[n * H + k + 1];
    }
  }

  v2f amat[16];
#pragma unroll
  for (int kc = 0; kc < 16; ++kc) {
    const int k = kc * 4 + 2 * kh;
    const float* hr = hid + (rowBase + m) * H + k;
    amat[kc].x = hr[0];
    amat[kc].y = hr[1];
  }

#pragma unroll
  for (int t = 0; t < 2; ++t) {
    const int n = t * 16 + m;
    const float bias = b_fc[n];
    v8f c;
#pragma unroll
    for (int j = 0; j < 8; ++j) c[j] = bias;
#pragma unroll
    for (int kc = 0; kc < 16; ++kc)
      c = wmma_f32_16x16x4(amat[kc], bmat[t][kc], c);
#pragma unroll
    for (int j = 0; j < 8; ++j) {
      const long r = rowBase + j + 8 * kh;
      // write-once, never re-read -> non-temporal store
      __builtin_nontemporal_store(c[j], &obs[r * 32 + n]);
    }
  }
}

// ---------------------------------------------------------------------------
extern "C" void kernel_launch(void* const* d_in, const int* in_sizes, int n_in,
                              void* d_out, int out_size, void* d_ws, size_t ws_size,
                              hipStream_t stream) {
  (void)in_sizes; (void)n_in; (void)out_size; (void)d_ws; (void)ws_size;
  const float* x    = (const float*)d_in[0];
  const float* W_ih = (const float*)d_in[1];
  const float* b_ih = (const float*)d_in[2];
  const float* W_hh = (const float*)d_in[3];
  const float* b_hh = (const float*)d_in[4];
  const float* W_fc = (const float*)d_in[5];
  const float* b_fc = (const float*)d_in[6];

  const long BP = 8L * 32, S = 2048, H = 64, O = 32;
  float* out    = (float*)d_out;
  float* obs    = out;                 // [BP, S, O]
  float* hid    = obs + BP * S * O;    // [BP, S, H]  (pre -> hidden, in place)
  float* h_last = hid + BP * S * H;    // [BP, H]

  // 524288 rows / (4 waves * 16 rows) = 8192 WGs for the big GEMMs
  pre_kernel<<<8192, 128, 0, stream>>>(x, W_ih, b_ih, b_hh, hid);
  rnn_scan_kernel<<<16, 128, 0, stream>>>(W_hh, hid, h_last);
  obs_kernel<<<8192, 128, 0, stream>>>(hid, W_fc, b_fc, obs);
}